// HouseholdAssignmentGNN_90829968376535
// MI455X (gfx1250) — compile-verified
//
#include <hip/hip_runtime.h>

// HouseholdAssignmentGNN for MI455X (gfx1250, wave32, WMMA)
//
// Pipeline:
//   zero(agg,cnt) -> scatter(x)   -> sage_update(w1) -> h
//   zero(agg,cnt) -> scatter(h)   -> sage_update(w2) -> h (in-place)
//   fc(h) -> out  (non-temporal stores, 400 MB stream)
//
// All GEMMs use V_WMMA_F32_16X16X4_F32 (fp32-exact, matches reference).

typedef __attribute__((ext_vector_type(2))) float v2f;
typedef __attribute__((ext_vector_type(8))) float v8f;

#define GNN_N_NODES 100000
#define GNN_D       64
#define GNN_N_HH    1000
#define GNN_NTILES  (GNN_N_NODES / 16)   // 6250 (exact)

// ---------------------------------------------------------------- zero
__global__ void gnn_zero_kernel(float* __restrict__ p, int n) {
    int i = blockIdx.x * blockDim.x + threadIdx.x;
    if (i < n) p[i] = 0.0f;
}

// ---------------------------------------------------------------- scatter-add (mean aggregation numerator + degree)
// 16 threads per edge; each handles 4 contiguous features (float4 gather, 4 f32 atomics).
// agg/cnt tables fit in L2 (25.6 MB + 0.4 MB), so atomics stay on-chip.
__global__ void gnn_scatter_kernel(const float* __restrict__ h,
                                   const int*   __restrict__ src,
                                   const int*   __restrict__ dst,
                                   float* __restrict__ agg,
                                   float* __restrict__ cnt,
                                   int E) {
    int t = blockIdx.x * blockDim.x + threadIdx.x;
    int e = t >> 4;
    if (e >= E) return;
    int c = (t & 15) << 2;                       // feature chunk base (0..60)
    int s = src[e];
    int d = dst[e];
    const float4 v = *reinterpret_cast<const float4*>(h + (size_t)s * GNN_D + c);
    float* a = agg + (size_t)d * GNN_D + c;
    atomicAdd(a + 0, v.x);
    atomicAdd(a + 1, v.y);
    atomicAdd(a + 2, v.z);
    atomicAdd(a + 3, v.w);
    if (c == 0) atomicAdd(cnt + d, 1.0f);
}

// ---------------------------------------------------------------- SAGE node update
// out[tile 16 nodes][64] = relu( (agg/max(cnt,1)) @ w_l^T + b_l + xin @ w_r^T )
// One wave per 16-node tile; 4 accumulator tiles cover all 64 output cols.
__global__ void __launch_bounds__(128)
gnn_sage_update_kernel(const float* __restrict__ agg,
                       const float* __restrict__ cnt,
                       const float* __restrict__ xin,
                       const float* __restrict__ w_l,
                       const float* __restrict__ b_l,
                       const float* __restrict__ w_r,
                       float* __restrict__ out) {
    int tile = blockIdx.x * blockDim.y + threadIdx.y;   // wave-uniform
    if (tile >= GNN_NTILES) return;                     // wave-uniform exit -> EXEC all-1 below

    int lane = threadIdx.x;
    int mrow = lane & 15;            // A: M index; B: N index within tile
    int kb   = (lane >> 4) << 1;     // 0 (lanes 0-15) or 2 (lanes 16-31)
    int node = tile * 16 + mrow;

    float inv = 1.0f / fmaxf(cnt[node], 1.0f);
    const float* arow = agg + (size_t)node * GNN_D;
    const float* xrow = xin + (size_t)node * GNN_D;

    v8f acc[4] = {};

    for (int k = 0; k < GNN_D; k += 4) {
        int kk = k + kb;
        v2f am, ax;
        am.x = arow[kk] * inv;
        am.y = arow[kk + 1] * inv;
        const float2 xv = *reinterpret_cast<const float2*>(xrow + kk);
        ax.x = xv.x;  ax.y = xv.y;
#pragma unroll
        for (int n = 0; n < 4; ++n) {
            const float2 blv = *reinterpret_cast<const float2*>(w_l + (size_t)(n * 16 + mrow) * GNN_D + kk);
            const float2 brv = *reinterpret_cast<const float2*>(w_r + (size_t)(n * 16 + mrow) * GNN_D + kk);
            v2f bl; bl.x = blv.x; bl.y = blv.y;
            v2f br; br.x = brv.x; br.y = brv.y;
            acc[n] = __builtin_amdgcn_wmma_f32_16x16x4_f32(false, am, false, bl,
                                                           (short)0, acc[n], false, false);
            acc[n] = __builtin_amdgcn_wmma_f32_16x16x4_f32(false, ax, false, br,
                                                           (short)0, acc[n], false, false);
        }
    }

    // C/D layout: VGPR v, lane L -> M = v + 8*(L>=16), N = L%16
    int mhalf = (lane >> 4) << 3;    // 0 or 8
#pragma unroll
    for (int n = 0; n < 4; ++n) {
        int col  = n * 16 + mrow;
        float bias = b_l[col];
#pragma unroll
        for (int v = 0; v < 8; ++v) {
            int onode = tile * 16 + v + mhalf;
            float val = acc[n][v] + bias;
            out[(size_t)onode * GNN_D + col] = val > 0.0f ? val : 0.0f;
        }
    }
}

// ---------------------------------------------------------------- final FC: out[N,1000] = h @ fc_w^T + fc_b
// One wave computes a 16x16 output tile; 63 column tiles (last is partial: cols 992..999).
__global__ void __launch_bounds__(128)
gnn_fc_kernel(const float* __restrict__ h,
              const float* __restrict__ fc_w,
              const float* __restrict__ fc_b,
              float* __restrict__ out) {
    int tile = blockIdx.x;                                  // node tile 0..6249
    int ct   = blockIdx.y * blockDim.y + threadIdx.y;       // col tile (wave-uniform)
    if (ct >= (GNN_N_HH + 15) / 16) return;                 // 63 tiles; wave-uniform exit

    int lane = threadIdx.x;
    int mrow = lane & 15;
    int kb   = (lane >> 4) << 1;
    int node = tile * 16 + mrow;
    int ncol = ct * 16 + mrow;
    int nclamp = ncol < GNN_N_HH ? ncol : (GNN_N_HH - 1);   // clamp load row; store is guarded

    const float* hrow = h    + (size_t)node   * GNN_D;
    const float* wrow = fc_w + (size_t)nclamp * GNN_D;

    v8f acc = {};
    for (int k = 0; k < GNN_D; k += 4) {
        int kk = k + kb;
        const float2 av = *reinterpret_cast<const float2*>(hrow + kk);
        const float2 bv = *reinterpret_cast<const float2*>(wrow + kk);
        v2f a; a.x = av.x; a.y = av.y;
        v2f b; b.x = bv.x; b.y = bv.y;
        acc = __builtin_amdgcn_wmma_f32_16x16x4_f32(false, a, false, b,
                                                    (short)0, acc, false, false);
    }

    int mhalf = (lane >> 4) << 3;
    if (ncol < GNN_N_HH) {          // per-lane divergence only AFTER all WMMA ops
        float bias = fc_b[ncol];
#pragma unroll
        for (int v = 0; v < 8; ++v) {
            int onode = tile * 16 + v + mhalf;
            // 400 MB streaming output > 192 MB L2: write non-temporal.
            __builtin_nontemporal_store(acc[v] + bias, out + (size_t)onode * GNN_N_HH + ncol);
        }
    }
}

// ---------------------------------------------------------------- launcher
extern "C" void kernel_launch(void* const* d_in, const int* in_sizes, int n_in,
                              void* d_out, int out_size, void* d_ws, size_t ws_size,
                              hipStream_t stream) {
    const float* x    = (const float*)d_in[0];
    const int*   edges= (const int*)  d_in[1];   // [2, E] (harness integer convention)
    const float* w1_l = (const float*)d_in[2];
    const float* b1_l = (const float*)d_in[3];
    const float* w1_r = (const float*)d_in[4];
    const float* w2_l = (const float*)d_in[5];
    const float* b2_l = (const float*)d_in[6];
    const float* w2_r = (const float*)d_in[7];
    const float* fc_w = (const float*)d_in[8];
    const float* fc_b = (const float*)d_in[9];
    float* out = (float*)d_out;

    const int E = in_sizes[1] / 2;
    const int* src = edges;
    const int* dst = edges + E;

    // Workspace layout (agg and cnt contiguous so one zero-kernel clears both):
    //   agg: N*64 f32 (25.6 MB)   cnt: N f32 (0.4 MB)   h: N*64 f32 (25.6 MB)
    float* agg = (float*)d_ws;
    float* cnt = agg + (size_t)GNN_N_NODES * GNN_D;
    float* h   = cnt + (size_t)GNN_N_NODES;

    const int ZN = GNN_N_NODES * GNN_D + GNN_N_NODES;     // 6,500,000
    const int sthreads = E * 16;                          // 25.6M

    dim3 zgrid((ZN + 255) / 256), zblk(256);
    dim3 sgrid((sthreads + 255) / 256), sblk(256);
    dim3 ugrid((GNN_NTILES + 3) / 4), ublk(32, 4);
    dim3 fgrid(GNN_NTILES, 16), fblk(32, 4);

    // ---- Layer 1 ----
    gnn_zero_kernel<<<zgrid, zblk, 0, stream>>>(agg, ZN);
    gnn_scatter_kernel<<<sgrid, sblk, 0, stream>>>(x, src, dst, agg, cnt, E);
    gnn_sage_update_kernel<<<ugrid, ublk, 0, stream>>>(agg, cnt, x, w1_l, b1_l, w1_r, h);

    // ---- Layer 2 (in-place over h: each wave only touches its own 16-row tile) ----
    gnn_zero_kernel<<<zgrid, zblk, 0, stream>>>(agg, ZN);
    gnn_scatter_kernel<<<sgrid, sblk, 0, stream>>>(h, src, dst, agg, cnt, E);
    gnn_sage_update_kernel<<<ugrid, ublk, 0, stream>>>(agg, cnt, h, w2_l, b2_l, w2_r, h);

    // ---- Final FC ----
    gnn_fc_kernel<<<fgrid, fblk, 0, stream>>>(h, fc_w, fc_b, out);
}